// CausalSelfAttention_36352603194175
// MI455X (gfx1250) — compile-verified
//
#include <hip/hip_runtime.h>

// CDNA5 / gfx1250 fused causal self-attention, fp32, V_WMMA_F32_16X16X4_F32.
// B=4096, T=64, DIM=128, NHEADS=4, HSIZE=32.
//
// Plan:
//   k0: transpose W_attn -> ws (384x128), W_proj -> ws (128x128)   [tiny]
//   k1: per-batch fused:  qkv = x@W+b (WMMA, into LDS, per head)
//                         scores = QK^T*scale + causal mask (WMMA)
//                         softmax (LDS), y = P@V (WMMA) -> d_out
//   k2: per-batch in-place: out = out @ W_proj + b_proj (WMMA, LDS-staged)

typedef __attribute__((ext_vector_type(2))) float v2f;
typedef __attribute__((ext_vector_type(8))) float v8f;

#define NBATCH 4096
#define TSEQ   64
#define DIM    128
#define NH     4
#define HS     32
#define N3     384

// Padded LDS strides (floats): even (keeps float2 8B-aligned), non-multiple of
// 64 (breaks bank conflicts for strided row access).
#define LQK 36    // 64x32 q/k tiles
#define LVT 66    // 32x64 transposed-V tile
#define LSS 68    // 64x64 score tile
#define LDX 132   // 64x128 proj tile

__device__ __forceinline__ v8f wmma_f32(v2f a, v2f b, v8f c) {
    // D(16x16,f32) = A(16x4,f32) x B(4x16,f32) + C
    return __builtin_amdgcn_wmma_f32_16x16x4_f32(
        /*neg_a=*/false, a, /*neg_b=*/false, b,
        /*c_mod=*/(short)0, c, /*reuse_a=*/false, /*reuse_b=*/false);
}

// ---------------------------------------------------------------------------
// Weight transpose: W[rows][cols] -> Wt[cols][rows]  (runs once per launch)
// ---------------------------------------------------------------------------
__global__ __launch_bounds__(256) void transpose_kernel(
    const float* __restrict__ W, float* __restrict__ Wt, int rows, int cols)
{
    int i = blockIdx.x * 256 + threadIdx.x;
    if (i < rows * cols) {
        int r = i / cols, c = i - r * cols;
        Wt[(size_t)c * rows + r] = W[i];
    }
}

// ---------------------------------------------------------------------------
// Fused QKV + attention. grid = B blocks, 128 threads (4 waves, wave = M-tile).
// wt = W_attn^T [384][128]; out = [B,T,DIM].
// ---------------------------------------------------------------------------
__global__ __launch_bounds__(128) void fused_attn_kernel(
    const float* __restrict__ x, const float* __restrict__ wt,
    const float* __restrict__ bias, float* __restrict__ out)
{
    __shared__ float qs [TSEQ * LQK];
    __shared__ float ksm[TSEQ * LQK];
    __shared__ float vt [HS   * LVT];   // V transposed: vt[hs][t]
    __shared__ float ss [TSEQ * LSS];

    const int b    = blockIdx.x;
    const int tid  = threadIdx.x;
    const int wave = tid >> 5;          // 0..3 == M-tile
    const int lane = tid & 31;
    const int half = lane >> 4;         // K-pair select: {0,1} vs {2,3}
    const int lr   = lane & 15;
    const int row  = wave * 16 + lr;    // t-row this lane supplies to A frags

    const float* xb = x   + (size_t)b * TSEQ * DIM;
    float*       ob = out + (size_t)b * TSEQ * DIM;

    // Hoisted x A-fragments: block reads x_b exactly once, reused for all
    // 4 heads x 6 column tiles of the QKV projection.
    v2f axf[32];
    #pragma unroll
    for (int k4 = 0; k4 < 32; ++k4)
        axf[k4] = *(const v2f*)(&xb[row * DIM + k4 * 4 + half * 2]);

    const float scale = 0.17677669529663687f;   // 1/sqrt(32)

    #pragma unroll 1
    for (int h = 0; h < NH; ++h) {
        // ---- QKV projection for head h: [64 x 96] (q|k|v, 32 cols each) ----
        #pragma unroll
        for (int nt = 0; nt < 6; ++nt) {
            const int reg   = nt >> 1;               // 0=q 1=k 2=v
            const int cloc  = (nt & 1) * 16 + lr;    // 0..31 within head
            const int cglob = reg * DIM + h * HS + cloc;
            const float* wc = wt + (size_t)cglob * DIM;
            v8f acc = {};
            #pragma unroll
            for (int k4 = 0; k4 < 32; ++k4) {
                v2f bf = *(const v2f*)(&wc[k4 * 4 + half * 2]);
                acc = wmma_f32(axf[k4], bf, acc);
            }
            const float bv = bias[cglob];
            #pragma unroll
            for (int i = 0; i < 8; ++i) {
                const int t = wave * 16 + half * 8 + i;
                const float v = acc[i] + bv;
                if (reg == 0)      qs [t * LQK + cloc] = v;
                else if (reg == 1) ksm[t * LQK + cloc] = v;
                else               vt [cloc * LVT + t] = v;   // store transposed
            }
        }
        __syncthreads();

        // ---- scores = (Q K^T) * scale, causal mask ----
        v2f aqf[8];
        #pragma unroll
        for (int k4 = 0; k4 < 8; ++k4)
            aqf[k4] = *(const v2f*)(&qs[row * LQK + k4 * 4 + half * 2]);
        #pragma unroll
        for (int nt = 0; nt < 4; ++nt) {
            const int col = nt * 16 + lr;
            v8f acc = {};
            #pragma unroll
            for (int k4 = 0; k4 < 8; ++k4) {
                // B = K^T: column n of B is row n of K -> contiguous float2
                v2f bf = *(const v2f*)(&ksm[col * LQK + k4 * 4 + half * 2]);
                acc = wmma_f32(aqf[k4], bf, acc);
            }
            #pragma unroll
            for (int i = 0; i < 8; ++i) {
                const int m = wave * 16 + half * 8 + i;
                ss[m * LSS + col] = (col <= m) ? acc[i] * scale : -1.0e30f;
            }
        }
        __syncthreads();

        // ---- row softmax (divergent, barrier-fenced) ----
        if (tid < TSEQ) {
            float* rowp = &ss[tid * LSS];
            float mx = rowp[0];
            for (int j = 1; j < TSEQ; ++j) mx = fmaxf(mx, rowp[j]);
            float sum = 0.0f;
            for (int j = 0; j < TSEQ; ++j) {
                float e = __expf(rowp[j] - mx); rowp[j] = e; sum += e;
            }
            const float inv = 1.0f / sum;
            for (int j = 0; j < TSEQ; ++j) rowp[j] *= inv;
        }
        __syncthreads();

        // ---- y = P[64x64] @ V[64x32]; B frags contiguous via transposed V ----
        v2f apf[16];
        #pragma unroll
        for (int k4 = 0; k4 < 16; ++k4)
            apf[k4] = *(const v2f*)(&ss[row * LSS + k4 * 4 + half * 2]);
        #pragma unroll
        for (int nt = 0; nt < 2; ++nt) {
            const int col = nt * 16 + lr;            // hs index
            v8f acc = {};
            #pragma unroll
            for (int k4 = 0; k4 < 16; ++k4) {
                v2f bf = *(const v2f*)(&vt[col * LVT + k4 * 4 + half * 2]);
                acc = wmma_f32(apf[k4], bf, acc);
            }
            #pragma unroll
            for (int i = 0; i < 8; ++i) {
                const int t = wave * 16 + half * 8 + i;
                ob[(size_t)t * DIM + h * HS + col] = acc[i];
            }
        }
        __syncthreads();   // qs/ksm/vt/ss reused by next head
    }
}

// ---------------------------------------------------------------------------
// In-place output projection: out_b = out_b @ W_proj + b_proj.
// grid = B blocks, 256 threads (8 waves). Tile staged in LDS (RAW-safe).
// wt = W_proj^T [128][128].
// ---------------------------------------------------------------------------
__global__ __launch_bounds__(256) void proj_kernel(
    float* __restrict__ io, const float* __restrict__ wt,
    const float* __restrict__ bias)
{
    __shared__ float ys[TSEQ * LDX];
    const int b   = blockIdx.x;
    const int tid = threadIdx.x;

    float* yb = io + (size_t)b * TSEQ * DIM;
    for (int i = tid; i < TSEQ * DIM; i += 256) {
        int r = i >> 7, c = i & 127;
        ys[r * LDX + c] = yb[i];
    }
    __syncthreads();   // all reads of the tile complete before any write

    const int wave = tid >> 5;
    const int lane = tid & 31;
    const int half = lane >> 4;
    const int lr   = lane & 15;
    const int mt   = wave & 3;
    const int row  = mt * 16 + lr;

    // Hoisted A fragments: one LDS read sweep per wave.
    v2f af[32];
    #pragma unroll
    for (int k4 = 0; k4 < 32; ++k4)
        af[k4] = *(const v2f*)(&ys[row * LDX + k4 * 4 + half * 2]);

    #pragma unroll
    for (int j = 0; j < 4; ++j) {
        const int nt  = (wave >> 2) + 2 * j;   // 0..7, disjoint across waves
        const int col = nt * 16 + lr;
        const float* wc = wt + (size_t)col * DIM;
        v8f acc = {};
        #pragma unroll
        for (int k4 = 0; k4 < 32; ++k4) {
            v2f bf = *(const v2f*)(&wc[k4 * 4 + half * 2]);
            acc = wmma_f32(af[k4], bf, acc);
        }
        const float bv = bias[col];
        #pragma unroll
        for (int i = 0; i < 8; ++i) {
            const int t = mt * 16 + half * 8 + i;
            yb[(size_t)t * DIM + col] = acc[i] + bv;
        }
    }
}

extern "C" void kernel_launch(void* const* d_in, const int* in_sizes, int n_in,
                              void* d_out, int out_size, void* d_ws, size_t ws_size,
                              hipStream_t stream) {
    // setup_inputs order: x, k_in, v_in, W_attn, b_attn, W_proj, b_proj
    const float* x      = (const float*)d_in[0];
    const float* W_attn = (const float*)d_in[3];
    const float* b_attn = (const float*)d_in[4];
    const float* W_proj = (const float*)d_in[5];
    const float* b_proj = (const float*)d_in[6];
    float* out = (float*)d_out;

    float* wt_attn = (float*)d_ws;              // [384][128]  (192 KB)
    float* wt_proj = wt_attn + (size_t)N3 * DIM; // [128][128]  (64 KB)

    transpose_kernel<<<(DIM * N3 + 255) / 256, 256, 0, stream>>>(W_attn, wt_attn, DIM, N3);
    transpose_kernel<<<(DIM * DIM + 255) / 256, 256, 0, stream>>>(W_proj, wt_proj, DIM, DIM);
    fused_attn_kernel<<<NBATCH, 128, 0, stream>>>(x, wt_attn, b_attn, out);
    proj_kernel     <<<NBATCH, 256, 0, stream>>>(out, wt_proj, b_proj);
}